// DecoderLayer_49185965473955
// MI455X (gfx1250) — compile-verified
//
#include <hip/hip_runtime.h>
#include <math.h>

#define D  256
#define H  8
#define HC 32

typedef __attribute__((ext_vector_type(2))) float v2f;
typedef __attribute__((ext_vector_type(8))) float v8f;

// ---- order-preserving float<->int encoding for atomic max on floats ----
__device__ __forceinline__ int ord_f32(float f) {
  int i = __float_as_int(f);
  return i >= 0 ? i : (i ^ 0x7FFFFFFF);
}
__device__ __forceinline__ float unord_f32(int i) {
  return __int_as_float(i >= 0 ? i : (i ^ 0x7FFFFFFF));
}

// ---------------- fill kernels ----------------
__global__ void fill_f32_kernel(float* p, float v, long n) {
  long i = (long)blockIdx.x * blockDim.x + threadIdx.x;
  if (i < n) p[i] = v;
}
__global__ void fill_i32_kernel(int* p, int v, long n) {
  long i = (long)blockIdx.x * blockDim.x + threadIdx.x;
  if (i < n) p[i] = v;
}

// ---------------- GEMM: Y[N,M] = act(X[N,256] @ W[256,M] + bias) ----------------
// One wave per 16(M-rows) x 64(N-cols) output tile, f32 WMMA 16x16x4, K fixed = 256.
__global__ void gemm_wmma_f32_kernel(const float* __restrict__ X,
                                     const float* __restrict__ W,
                                     const float* __restrict__ bias,
                                     float* __restrict__ Y,
                                     int Nrows, int Mcols, int act) {
  const int K = D;
  int wave = (int)(((long)blockIdx.x * blockDim.x + threadIdx.x) >> 5);
  int lane = threadIdx.x & 31;
  int ntiles = Mcols >> 6;             // 64-wide tiles
  int mtiles = (Nrows + 15) >> 4;
  int mt = wave / ntiles;
  int nt = wave - mt * ntiles;
  if (mt >= mtiles) return;            // wave-uniform exit (EXEC stays all-ones for WMMA)
  int half = lane >> 4;                // 0: K pair {0,1} | 1: K pair {2,3}
  int l15  = lane & 15;
  int row  = mt * 16 + l15;
  int rowc = row < Nrows ? row : Nrows - 1;
  const float* xp    = X + (long)rowc * K + half * 2;
  int col0 = nt * 64 + l15;
  const float* wbase = W + (long)(half * 2) * Mcols + col0;

  v8f acc[4] = {};
  #pragma unroll 2
  for (int k = 0; k < K; k += 4) {
    // A 16x4 frag: lane l<16 holds A[l][0..1], lane l+16 holds A[l][2..3] (contiguous f2)
    v2f a = *(const v2f*)(xp + k);
    const float* wp = wbase + (long)k * Mcols;
    // B 4x16 frags: VGPR r -> rows r and r+2 (row-striped across lanes)
    v2f b0, b1, b2, b3;
    b0[0] = wp[0];  b0[1] = wp[Mcols + 0];
    b1[0] = wp[16]; b1[1] = wp[Mcols + 16];
    b2[0] = wp[32]; b2[1] = wp[Mcols + 32];
    b3[0] = wp[48]; b3[1] = wp[Mcols + 48];
    acc[0] = __builtin_amdgcn_wmma_f32_16x16x4_f32(false, a, false, b0, (short)0, acc[0], false, false);
    acc[1] = __builtin_amdgcn_wmma_f32_16x16x4_f32(false, a, false, b1, (short)0, acc[1], false, false);
    acc[2] = __builtin_amdgcn_wmma_f32_16x16x4_f32(false, a, false, b2, (short)0, acc[2], false, false);
    acc[3] = __builtin_amdgcn_wmma_f32_16x16x4_f32(false, a, false, b3, (short)0, acc[3], false, false);
  }
  // C/D layout: VGPR r -> M=r (lanes 0-15), M=r+8 (lanes 16-31); N = lane&15
  #pragma unroll
  for (int t = 0; t < 4; ++t) {
    int col = col0 + t * 16;
    float bv = bias[col];
    #pragma unroll
    for (int r = 0; r < 8; ++r) {
      int orow = mt * 16 + half * 8 + r;
      if (orow < Nrows) {
        float x = acc[t][r] + bv;
        if (act) x = 0.5f * x * (1.0f + erff(x * 0.70710678118654752f));  // exact GELU
        Y[(long)orow * Mcols + col] = x;
      }
    }
  }
}

// ---------------- per-edge QK dot + segment max ----------------
// one wave per edge; 4 lanes per head, 8 dims per lane
__global__ void edge_qk_max_kernel(const float* __restrict__ Q,
                                   const float* __restrict__ KV,   // [n,512], K at +0
                                   const int* __restrict__ eq,
                                   const int* __restrict__ ek,
                                   float* __restrict__ qkbuf,      // [E,H]
                                   int* __restrict__ mbuf,         // [n_q,H] encoded
                                   int E) {
  long gid = (long)blockIdx.x * blockDim.x + threadIdx.x;
  int e = (int)(gid >> 5);
  if (e >= E) return;
  int lane = (int)(gid & 31);
  int h = lane >> 2, j = lane & 3;
  int q = eq[e], k = ek[e];
  const float4* qp = (const float4*)(Q  + (long)q * D       + h * HC + j * 8);
  const float4* kp = (const float4*)(KV + (long)k * (2 * D) + h * HC + j * 8);
  float4 q0 = qp[0], q1 = qp[1];
  float4 k0 = kp[0], k1 = kp[1];
  float p = q0.x * k0.x + q0.y * k0.y + q0.z * k0.z + q0.w * k0.w
          + q1.x * k1.x + q1.y * k1.y + q1.z * k1.z + q1.w * k1.w;
  p += __shfl_xor(p, 1);
  p += __shfl_xor(p, 2);
  if (j == 0) {
    float s = p * 0.17677669529663687f;  // 1/sqrt(32)
    qkbuf[(long)e * H + h] = s;
    atomicMax(&mbuf[(long)q * H + h], ord_f32(s));
  }
}

// ---------------- exp(qk - m) and segment sum ----------------
__global__ void edge_expsum_kernel(float* __restrict__ qkbuf,
                                   const int* __restrict__ eq,
                                   const int* __restrict__ mbuf,
                                   float* __restrict__ sbuf,
                                   int E) {
  long gid = (long)blockIdx.x * blockDim.x + threadIdx.x;
  int e = (int)(gid >> 3);
  if (e >= E) return;
  int h = (int)(gid & 7);
  int q = eq[e];
  float m  = unord_f32(mbuf[(long)q * H + h]);
  float ex = expf(qkbuf[(long)e * H + h] - m);
  qkbuf[(long)e * H + h] = ex;
  atomicAdd(&sbuf[(long)q * H + h], ex);
}

// ---------------- alpha * V scatter-accumulate ----------------
__global__ void edge_scatter_kernel(const float* __restrict__ qkbuf,
                                    const int* __restrict__ eq,
                                    const int* __restrict__ ek,
                                    const float* __restrict__ sbuf,
                                    const float* __restrict__ KV,   // V at +256
                                    float* __restrict__ acc,        // [n_q,256]
                                    int E) {
  long gid = (long)blockIdx.x * blockDim.x + threadIdx.x;
  int e = (int)(gid >> 5);
  if (e >= E) return;
  int lane = (int)(gid & 31);
  int h = lane >> 2, j = lane & 3;
  int q = eq[e], k = ek[e];
  float alpha = qkbuf[(long)e * H + h] / sbuf[(long)q * H + h];
  const float* vp = KV + (long)k * (2 * D) + D + h * HC + j * 8;
  float*       ap = acc + (long)q * D         + h * HC + j * 8;
  #pragma unroll
  for (int d = 0; d < 8; ++d) atomicAdd(&ap[d], alpha * vp[d]);
}

// ---------------- out = LayerNorm(A + B) * g + beta, one 256-thread block per row ----------------
__global__ void add_ln_kernel(const float* __restrict__ A,
                              const float* __restrict__ B,
                              const float* __restrict__ g,
                              const float* __restrict__ be,
                              float* __restrict__ out,
                              int rows) {
  int row = blockIdx.x;
  if (row >= rows) return;
  int tid = threadIdx.x;
  __shared__ float red[8];
  long base = (long)row * D;
  float x = A[base + tid] + B[base + tid];
  float s = x;
  #pragma unroll
  for (int o = 16; o > 0; o >>= 1) s += __shfl_down(s, o);
  if ((tid & 31) == 0) red[tid >> 5] = s;
  __syncthreads();
  float mean = (red[0] + red[1] + red[2] + red[3] + red[4] + red[5] + red[6] + red[7]) * (1.0f / D);
  float dlt = x - mean;
  __syncthreads();
  float s2 = dlt * dlt;
  #pragma unroll
  for (int o = 16; o > 0; o >>= 1) s2 += __shfl_down(s2, o);
  if ((tid & 31) == 0) red[tid >> 5] = s2;
  __syncthreads();
  float var = (red[0] + red[1] + red[2] + red[3] + red[4] + red[5] + red[6] + red[7]) * (1.0f / D);
  float r = rsqrtf(var + 1e-5f);
  out[base + tid] = dlt * r * g[tid] + be[tid];
}

// ---------------- host-side helpers ----------------
static inline dim3 grid_for(long threads) { return dim3((unsigned)((threads + 255) / 256)); }

static inline void launch_gemm(const float* X, const float* W, const float* b, float* Y,
                               int N, int M, int act, hipStream_t s) {
  long waves = (long)((N + 15) / 16) * (long)(M / 64);
  gemm_wmma_f32_kernel<<<grid_for(waves * 32), dim3(256), 0, s>>>(X, W, b, Y, N, M, act);
}

extern "C" void kernel_launch(void* const* d_in, const int* in_sizes, int n_in,
                              void* d_out, int out_size, void* d_ws, size_t ws_size,
                              hipStream_t stream) {
  const float* v    = (const float*)d_in[0];
  const float* c    = (const float*)d_in[1];
  const int*   adjp = (const int*)d_in[2];
  const int*   adjn = (const int*)d_in[3];
  const float* Wq   = (const float*)d_in[4];
  const float* bq   = (const float*)d_in[5];
  const float* Wkv  = (const float*)d_in[6];
  const float* bkv  = (const float*)d_in[7];
  const float* fvw1 = (const float*)d_in[8];
  const float* fvb1 = (const float*)d_in[9];
  const float* fvw2 = (const float*)d_in[10];
  const float* fvb2 = (const float*)d_in[11];
  const float* fcw1 = (const float*)d_in[12];
  const float* fcb1 = (const float*)d_in[13];
  const float* fcw2 = (const float*)d_in[14];
  const float* fcb2 = (const float*)d_in[15];
  const float* lavg = (const float*)d_in[16];
  const float* lavb = (const float*)d_in[17];
  const float* lfvg = (const float*)d_in[18];
  const float* lfvb = (const float*)d_in[19];
  const float* lacg = (const float*)d_in[20];
  const float* lacb = (const float*)d_in[21];
  const float* lfcg = (const float*)d_in[22];
  const float* lfcb = (const float*)d_in[23];

  const int NV = in_sizes[0] / D;
  const int NC = in_sizes[1] / D;
  const int Ep = in_sizes[2] / 2;
  const int En = in_sizes[3] / 2;
  const int* ep_c = adjp;       // adj row0: clause indices
  const int* ep_v = adjp + Ep;  // adj row1: variable indices
  const int* en_c = adjn;
  const int* en_v = adjn + En;

  // ---- workspace layout (floats) ----
  float* ws = (float*)d_ws;
  size_t off = 0;
  float* Qv  = ws + off; off += (size_t)NV * D;
  float* KVv = ws + off; off += (size_t)NV * 2 * D;   // later reused as FFN hidden (NC*256 == NV*512)
  float* Qc  = ws + off; off += (size_t)NC * D;       // later reused as FFN-2 output (c)
  float* KVc = ws + off; off += (size_t)NC * 2 * D;
  int*   m_pv = (int*)(ws + off); off += (size_t)NV * H;
  int*   m_nv = (int*)(ws + off); off += (size_t)NV * H;
  int*   m_pc = (int*)(ws + off); off += (size_t)NC * H;
  int*   m_nc = (int*)(ws + off); off += (size_t)NC * H;
  float* s_pv = ws + off; off += (size_t)NV * H;
  float* s_nv = ws + off; off += (size_t)NV * H;
  float* s_pc = ws + off; off += (size_t)NC * H;
  float* s_nc = ws + off; off += (size_t)NC * H;
  float* qk_pv = ws + off; off += (size_t)Ep * H;
  float* qk_nv = ws + off; off += (size_t)En * H;
  float* qk_pc = ws + off; off += (size_t)Ep * H;
  float* qk_nc = ws + off; off += (size_t)En * H;

  float* out   = (float*)d_out;
  float* acc_v = out;                     // attention accumulators live in d_out
  float* acc_c = out + (size_t)NV * D;
  float* hbuf  = KVv;                     // FFN hidden scratch (after attention done)
  float* y2v   = Qv;                      // FFN-2 outputs
  float* y2c   = Qc;

  const long m_total = (long)2 * (NV + NC) * H;

  // ---- 1) shared projections (computed once; reference recomputes per polarity) ----
  launch_gemm(v, Wq,  bq,  Qv,  NV, D,     0, stream);
  launch_gemm(v, Wkv, bkv, KVv, NV, 2 * D, 0, stream);
  launch_gemm(c, Wq,  bq,  Qc,  NC, D,     0, stream);
  launch_gemm(c, Wkv, bkv, KVc, NC, 2 * D, 0, stream);

  // ---- 2) init softmax state + accumulators (every call: deterministic) ----
  fill_i32_kernel<<<grid_for(m_total), 256, 0, stream>>>(m_pv, (int)0x80000000, m_total);
  fill_f32_kernel<<<grid_for(m_total), 256, 0, stream>>>(s_pv, 0.0f, m_total);
  fill_f32_kernel<<<grid_for((long)(NV + NC) * D), 256, 0, stream>>>(out, 0.0f, (long)(NV + NC) * D);

  // ---- 3) per-edge QK + segment max ----
  edge_qk_max_kernel<<<grid_for((long)Ep * 32), 256, 0, stream>>>(Qv, KVc, ep_v, ep_c, qk_pv, m_pv, Ep);
  edge_qk_max_kernel<<<grid_for((long)Ep * 32), 256, 0, stream>>>(Qc, KVv, ep_c, ep_v, qk_pc, m_pc, Ep);
  edge_qk_max_kernel<<<grid_for((long)En * 32), 256, 0, stream>>>(Qv, KVc, en_v, en_c, qk_nv, m_nv, En);
  edge_qk_max_kernel<<<grid_for((long)En * 32), 256, 0, stream>>>(Qc, KVv, en_c, en_v, qk_nc, m_nc, En);

  // ---- 4) exp + segment sum ----
  edge_expsum_kernel<<<grid_for((long)Ep * 8), 256, 0, stream>>>(qk_pv, ep_v, m_pv, s_pv, Ep);
  edge_expsum_kernel<<<grid_for((long)Ep * 8), 256, 0, stream>>>(qk_pc, ep_c, m_pc, s_pc, Ep);
  edge_expsum_kernel<<<grid_for((long)En * 8), 256, 0, stream>>>(qk_nv, en_v, m_nv, s_nv, En);
  edge_expsum_kernel<<<grid_for((long)En * 8), 256, 0, stream>>>(qk_nc, en_c, m_nc, s_nc, En);

  // ---- 5) alpha*V scatter (pos+neg share accumulator; residual sums them anyway) ----
  edge_scatter_kernel<<<grid_for((long)Ep * 32), 256, 0, stream>>>(qk_pv, ep_v, ep_c, s_pv, KVc, acc_v, Ep);
  edge_scatter_kernel<<<grid_for((long)Ep * 32), 256, 0, stream>>>(qk_pc, ep_c, ep_v, s_pc, KVv, acc_c, Ep);
  edge_scatter_kernel<<<grid_for((long)En * 32), 256, 0, stream>>>(qk_nv, en_v, en_c, s_nv, KVc, acc_v, En);
  edge_scatter_kernel<<<grid_for((long)En * 32), 256, 0, stream>>>(qk_nc, en_c, en_v, s_nc, KVv, acc_c, En);

  // ---- 6) residual + LN (attention) ----
  add_ln_kernel<<<dim3(NV), dim3(D), 0, stream>>>(v, acc_v, lavg, lavb, acc_v, NV);
  add_ln_kernel<<<dim3(NC), dim3(D), 0, stream>>>(c, acc_c, lacg, lacb, acc_c, NC);

  // ---- 7) FFN v: gelu(v1@W1+b1)@W2+b2, then residual + LN ----
  launch_gemm(acc_v, fvw1, fvb1, hbuf, NV, D, 1, stream);
  launch_gemm(hbuf,  fvw2, fvb2, y2v,  NV, D, 0, stream);
  add_ln_kernel<<<dim3(NV), dim3(D), 0, stream>>>(acc_v, y2v, lfvg, lfvb, acc_v, NV);

  // ---- 8) FFN c ----
  launch_gemm(acc_c, fcw1, fcb1, hbuf, NC, D, 1, stream);
  launch_gemm(hbuf,  fcw2, fcb2, y2c,  NC, D, 0, stream);
  add_ln_kernel<<<dim3(NC), dim3(D), 0, stream>>>(acc_c, y2c, lfcg, lfcb, acc_c, NC);

  (void)n_in; (void)out_size; (void)ws_size;
}